// RmiModel_22419729286023
// MI455X (gfx1250) — compile-verified
//
#include <hip/hip_runtime.h>
#include <hip/hip_bf16.h>
#include <math.h>

#define T_LEN   1024
#define N_STEP  (T_LEN - 1)   // 1023 scan steps
#define NTHR    256
#define GSTRIDE 20            // floats per tree slot (16 used, padded for 16B-aligned float4 ops)

#ifdef __has_builtin
#  if __has_builtin(__builtin_amdgcn_global_load_async_to_lds_b128)
#    define HAVE_ASYNC_B128 1
#  endif
#  if __has_builtin(__builtin_amdgcn_s_wait_asynccnt)
#    define HAVE_WAIT_ASYNC 1
#  endif
#endif

// 16-byte GCC-style int vector, matching the builtin's parameter type
typedef int v4i __attribute__((vector_size(4 * sizeof(int))));
typedef __attribute__((address_space(1))) v4i* gptr_v4i;
typedef __attribute__((address_space(3))) v4i* lptr_v4i;

// Group element of the preintegration group:
//   G = [[C, v, r],[0,1,tau],[0,0,1]]   (5x5, C is 3x3 row-major)
// f[0..8]=C, f[9..11]=v, f[12..14]=r, f[15]=tau
struct G16 { float f[16]; };

__device__ __forceinline__ G16 g_compose(const G16& a, const G16& b) {
    G16 o;
#pragma unroll
    for (int i = 0; i < 3; ++i) {
        const float a0 = a.f[3*i+0], a1 = a.f[3*i+1], a2 = a.f[3*i+2];
        o.f[3*i+0] = a0*b.f[0] + a1*b.f[3] + a2*b.f[6];
        o.f[3*i+1] = a0*b.f[1] + a1*b.f[4] + a2*b.f[7];
        o.f[3*i+2] = a0*b.f[2] + a1*b.f[5] + a2*b.f[8];
        o.f[9+i]  = a0*b.f[9]  + a1*b.f[10] + a2*b.f[11] + a.f[9+i];
        o.f[12+i] = a0*b.f[12] + a1*b.f[13] + a2*b.f[14] + a.f[9+i]*b.f[15] + a.f[12+i];
    }
    o.f[15] = a.f[15] + b.f[15];
    return o;
}

__device__ __forceinline__ void g_store(float* base, const G16& g) {
#pragma unroll
    for (int q = 0; q < 4; ++q)
        ((float4*)base)[q] = make_float4(g.f[4*q], g.f[4*q+1], g.f[4*q+2], g.f[4*q+3]);
}

__device__ __forceinline__ G16 g_load(const float* base) {
    G16 g;
#pragma unroll
    for (int q = 0; q < 4; ++q) {
        float4 u = ((const float4*)base)[q];
        g.f[4*q+0] = u.x; g.f[4*q+1] = u.y; g.f[4*q+2] = u.z; g.f[4*q+3] = u.w;
    }
    return g;
}

// One scan step as a group element: C = exp(skew(w*dt)), v = a*dt, r = 0.5*a*dt^2, tau = dt.
__device__ __forceinline__ G16 make_leaf(float dt, float wx, float wy, float wz,
                                         float ax, float ay, float az) {
    const float px = wx * dt, py = wy * dt, pz = wz * dt;
    const float th2 = px*px + py*py + pz*pz;
    const bool  sm  = th2 < 1e-8f;
    const float th2s = sm ? 1.0f : th2;
    const float th   = sqrtf(th2s);
    const float A  = sm ? (1.0f - th2 * (1.0f/6.0f))  : (sinf(th) / th);
    const float Bc = sm ? (0.5f - th2 * (1.0f/24.0f)) : ((1.0f - cosf(th)) / th2s);
    const float xx = px*px, yy = py*py, zz = pz*pz;
    const float xy = px*py, xz = px*pz, yz = py*pz;
    G16 g;
    g.f[0] = 1.0f - Bc*(yy+zz); g.f[1] = -A*pz + Bc*xy;     g.f[2] =  A*py + Bc*xz;
    g.f[3] =  A*pz + Bc*xy;     g.f[4] = 1.0f - Bc*(xx+zz); g.f[5] = -A*px + Bc*yz;
    g.f[6] = -A*py + Bc*xz;     g.f[7] =  A*px + Bc*yz;     g.f[8] = 1.0f - Bc*(xx+yy);
    g.f[9]  = ax*dt; g.f[10] = ay*dt; g.f[11] = az*dt;
    const float h = 0.5f*dt*dt;
    g.f[12] = ax*h;  g.f[13] = ay*h;  g.f[14] = az*h;
    g.f[15] = dt;
    return g;
}

__global__ __launch_bounds__(NTHR)
void rmi_preint_kernel(const float* __restrict__ x, float* __restrict__ out) {
    __shared__ __align__(16) float s_x[7 * T_LEN];          // 28 KB staged batch tile
    __shared__ __align__(16) float s_g[NTHR * GSTRIDE];     // 20 KB reduction tree

    const int t = threadIdx.x;
    const int b = blockIdx.x;
    const float* gx = x + (size_t)b * 7 * T_LEN;

    // ---- Stage batch tile global -> LDS (coalesced async b128 per lane) ----
#if HAVE_ASYNC_B128
#pragma unroll
    for (int i = 0; i < 7; ++i) {
        const int e = (i * NTHR + t) * 4;   // float index, 16B aligned
        __builtin_amdgcn_global_load_async_to_lds_b128(
            (gptr_v4i)(gx + e),
            (lptr_v4i)(s_x + e),
            0, 0);
    }
#  if HAVE_WAIT_ASYNC
    __builtin_amdgcn_s_wait_asynccnt(0);
#  else
    asm volatile("s_wait_asynccnt 0" ::: "memory");
#  endif
#else
#pragma unroll
    for (int i = 0; i < 7; ++i) {
        const int e = (i * NTHR + t) * 4;
        *(float4*)(s_x + e) = *(const float4*)(gx + e);
    }
#endif
    __syncthreads();

    // ---- Each thread builds 4 leaves [4t, 4t+4) and composes them in order ----
    float tv[5];
    {
        float4 u = *(const float4*)(s_x + 4*t);
        tv[0] = u.x; tv[1] = u.y; tv[2] = u.z; tv[3] = u.w;
        tv[4] = (t < NTHR - 1) ? s_x[4*t + 4] : 0.0f;   // last thread's 4th leaf is skipped
    }
    float wv[3][4], av[3][4];
#pragma unroll
    for (int c = 0; c < 3; ++c) {
        float4 u = *(const float4*)(s_x + (1 + c) * T_LEN + 4*t);
        wv[c][0] = u.x; wv[c][1] = u.y; wv[c][2] = u.z; wv[c][3] = u.w;
        float4 q = *(const float4*)(s_x + (4 + c) * T_LEN + 4*t);
        av[c][0] = q.x; av[c][1] = q.y; av[c][2] = q.z; av[c][3] = q.w;
    }

    G16 g;
    bool first = true;
#pragma unroll
    for (int j = 0; j < 4; ++j) {
        const int k = 4*t + j;
        if (k < N_STEP) {
            const float dt = tv[j+1] - tv[j];
            G16 l = make_leaf(dt, wv[0][j], wv[1][j], wv[2][j],
                                  av[0][j], av[1][j], av[2][j]);
            g = first ? l : g_compose(g, l);
            first = false;
        }
    }

    // ---- 8-level tree reduction over 256 slots (left operand = earlier steps) ----
    g_store(s_g + t * GSTRIDE, g);
#pragma unroll
    for (int s = 1; s < NTHR; s <<= 1) {
        __syncthreads();
        if ((t & (2*s - 1)) == 0) {
            G16 h = g_load(s_g + (t + s) * GSTRIDE);
            g = g_compose(g, h);
            g_store(s_g + t * GSTRIDE, g);
        }
    }

    if (t == 0) {
        float* o = out + (size_t)b * 15;
#pragma unroll
        for (int i = 0; i < 9; ++i) o[i] = g.f[i];        // DC row-major
#pragma unroll
        for (int i = 0; i < 3; ++i) o[9 + i] = g.f[9 + i];   // DV
#pragma unroll
        for (int i = 0; i < 3; ++i) o[12 + i] = g.f[12 + i]; // DR
    }
}

extern "C" void kernel_launch(void* const* d_in, const int* in_sizes, int n_in,
                              void* d_out, int out_size, void* d_ws, size_t ws_size,
                              hipStream_t stream) {
    const float* x = (const float*)d_in[0];
    float* out = (float*)d_out;
    const int B = in_sizes[0] / (7 * T_LEN);   // 4096
    rmi_preint_kernel<<<B, NTHR, 0, stream>>>(x, out);
}